// RefineDet_12713103197200
// MI455X (gfx1250) — compile-verified
//
#include <hip/hip_runtime.h>

// RefineDet detect post-process for MI455X (gfx1250, wave32).
// Pipeline: zero-out -> decode+transpose -> radix top-K select -> WMMA-assisted NMS.

#define BATCH   32
#define P_NUM   16320
#define NCLS    21
#define TOPK    500
#define KPAD    512
#define CONF_TH 0.01f
#define NMS_TH  0.45f
#define OBJ_TH  0.01f

typedef __attribute__((ext_vector_type(2))) float v2f;
typedef __attribute__((ext_vector_type(8))) float v8f;

__device__ __forceinline__ unsigned fbits(float f) { return __float_as_uint(f); }

// ---------------------------------------------------------------- zero output
__global__ void k_zero(float* __restrict__ out, int n) {
    int i = blockIdx.x * 256 + threadIdx.x;
    if (i < n) out[i] = 0.f;
}

// ------------------------------------------------- decode + conf transposition
// default = center_size(decode(arm_loc, priors)) simplifies to
// (prior.cxy + 0.1*al.xy*prior.wh, prior.wh*exp(0.2*al.zw)); then second decode.
__global__ void k_decode(const float* __restrict__ arm_loc,
                         const float* __restrict__ arm_conf,
                         const float* __restrict__ odm_loc,
                         const float* __restrict__ odm_conf,
                         const float* __restrict__ priors,
                         float* __restrict__ boxes,
                         float* __restrict__ conf_t) {
    int idx = blockIdx.x * blockDim.x + threadIdx.x;
    if (idx >= BATCH * P_NUM) return;
    int b = idx / P_NUM, p = idx - b * P_NUM;

    float4 pr = reinterpret_cast<const float4*>(priors)[p];
    float4 al = reinterpret_cast<const float4*>(arm_loc)[idx];
    float4 ol = reinterpret_cast<const float4*>(odm_loc)[idx];

    float dcx = pr.x + al.x * 0.1f * pr.z;
    float dcy = pr.y + al.y * 0.1f * pr.w;
    float dw  = pr.z * expf(0.2f * al.z);
    float dh  = pr.w * expf(0.2f * al.w);

    float cx = dcx + ol.x * 0.1f * dw;
    float cy = dcy + ol.y * 0.1f * dh;
    float w  = dw * expf(0.2f * ol.z);
    float h  = dh * expf(0.2f * ol.w);

    reinterpret_cast<float4*>(boxes)[idx] =
        make_float4(cx - 0.5f * w, cy - 0.5f * h, cx + 0.5f * w, cy + 0.5f * h);

    float obj  = arm_conf[(size_t)idx * 2 + 1];
    bool  keep = obj > OBJ_TH;
    const float* oc = odm_conf + (size_t)idx * NCLS;
    for (int c = 1; c < NCLS; ++c)
        conf_t[((size_t)(b * 20 + (c - 1))) * P_NUM + p] = keep ? oc[c] : 0.f;
}

// --------------------------------------------- exact top-K via 8-bit radix select
// One workgroup per (image,class). Stable tie-break (lowest prior index) to match
// jax.lax.top_k ordering semantics used by the greedy argmax.
__global__ void k_select(const float* __restrict__ conf_t,
                         const float* __restrict__ boxes,
                         float* __restrict__ cand_score,
                         unsigned* __restrict__ cand_idx,
                         float* __restrict__ cand_box) {
    __shared__ unsigned hist[256];
    __shared__ unsigned chunkCnt[256];
    __shared__ unsigned chunkBase[256];
    __shared__ unsigned s_nValid, s_prefix, s_m, s_cnt;

    int t  = threadIdx.x;
    int bc = blockIdx.x;
    int b  = bc / 20;
    const float* sc = conf_t + (size_t)bc * P_NUM;

    if (t == 0) { s_nValid = 0; s_cnt = 0; s_prefix = 0; s_m = TOPK; }
    __syncthreads();

    unsigned lc = 0;
    for (int p = t; p < P_NUM; p += 256)
        if (sc[p] > CONF_TH) lc++;
    atomicAdd(&s_nValid, lc);
    __syncthreads();
    unsigned nValid = s_nValid;

    unsigned T, mEq, gtBase;
    if (nValid <= TOPK) {
        T = fbits(CONF_TH); mEq = 0; gtBase = 0;       // take everything > CONF_TH
    } else {
        for (int shift = 24; shift >= 0; shift -= 8) {
            hist[t] = 0;
            __syncthreads();
            unsigned prefix = s_prefix;
            unsigned mask = (shift == 24) ? 0u : (0xFFFFFFFFu << (shift + 8));
            for (int p = t; p < P_NUM; p += 256) {
                float s = sc[p];
                if (s > CONF_TH) {
                    unsigned u = fbits(s);              // positives: monotonic bits
                    if ((u & mask) == prefix)
                        atomicAdd(&hist[(u >> shift) & 0xFF], 1u);
                }
            }
            __syncthreads();
            if (t == 0) {
                unsigned m = s_m, cum = 0; int chosen = 0;
                for (int bkt = 255; bkt >= 0; --bkt) {
                    if (cum + hist[bkt] >= m) { chosen = bkt; m -= cum; break; }
                    cum += hist[bkt];
                }
                s_m = m;
                s_prefix = prefix | ((unsigned)chosen << shift);
            }
            __syncthreads();
        }
        T = s_prefix; mEq = s_m; gtBase = TOPK - mEq;   // count(u > T) == TOPK - mEq
    }

    float*    cs = cand_score + (size_t)bc * KPAD;
    unsigned* ci = cand_idx   + (size_t)bc * KPAD;
    float*    cb = cand_box   + (size_t)bc * KPAD * 4;
    for (int s = t; s < KPAD; s += 256) {
        cs[s] = 0.f; ci[s] = 0u;
        cb[s * 4] = 0.f; cb[s * 4 + 1] = 0.f; cb[s * 4 + 2] = 0.f; cb[s * 4 + 3] = 0.f;
    }
    __syncthreads();

    const float* bxs = boxes + (size_t)b * P_NUM * 4;
    // strictly-greater candidates: order-free (greedy argmax keys restore order)
    for (int p = t; p < P_NUM; p += 256) {
        float s = sc[p];
        if (s > CONF_TH && fbits(s) > T) {
            unsigned pos = atomicAdd(&s_cnt, 1u);
            cs[pos] = s; ci[pos] = (unsigned)p;
            cb[pos * 4 + 0] = bxs[p * 4 + 0];
            cb[pos * 4 + 1] = bxs[p * 4 + 1];
            cb[pos * 4 + 2] = bxs[p * 4 + 2];
            cb[pos * 4 + 3] = bxs[p * 4 + 3];
        }
    }
    // equals-to-threshold: take the mEq lowest prior indices (stable top_k tie rule)
    if (mEq > 0) {
        int start = t * 64; unsigned cnt = 0;
        for (int k = 0; k < 64; ++k) {
            int p = start + k;
            if (p < P_NUM) { float s = sc[p]; if (s > CONF_TH && fbits(s) == T) cnt++; }
        }
        chunkCnt[t] = cnt;
        __syncthreads();
        if (t == 0) {
            unsigned run = 0;
            for (int i = 0; i < 256; ++i) { chunkBase[i] = run; run += chunkCnt[i]; }
        }
        __syncthreads();
        unsigned run = chunkBase[t];
        for (int k = 0; k < 64; ++k) {
            int p = start + k;
            if (p >= P_NUM) break;
            float s = sc[p];
            if (s > CONF_TH && fbits(s) == T) {
                if (run < mEq) {
                    unsigned pos = gtBase + run;
                    cs[pos] = s; ci[pos] = (unsigned)p;
                    cb[pos * 4 + 0] = bxs[p * 4 + 0];
                    cb[pos * 4 + 1] = bxs[p * 4 + 1];
                    cb[pos * 4 + 2] = bxs[p * 4 + 2];
                    cb[pos * 4 + 3] = bxs[p * 4 + 3];
                }
                run++;
            }
        }
    }
}

// ------------------------------------------------------------ WMMA-assisted NMS
// Phase 1: 512x512 suppression bit-matrix in LDS. The union term area_i+area_j is
// a rank-2 outer product computed on the matrix pipe: D = [area_i,1]x[1;area_j]
// via V_WMMA_F32_16X16X4_F32; VALU supplies min/max intersection + threshold test.
// Phase 2: single wave32 greedy scan over register-resident (score,~idx) keys.
__global__ void k_nms(const float* __restrict__ cand_score,
                      const unsigned* __restrict__ cand_idx,
                      const float* __restrict__ cand_box,
                      float* __restrict__ out) {
    __shared__ float  areaArr[KPAD];
    __shared__ float4 candBoxL[KPAD];
    __shared__ unsigned supMat[KPAD * 16];       // 512 rows x 512 bits = 32 KB
    __shared__ unsigned long long redK[32];
    __shared__ int redS[32];
    __shared__ unsigned long long gKey;
    __shared__ int gSlot;

    int t = threadIdx.x, bc = blockIdx.x;
    int b = bc / 20, cl = bc % 20;

    const float*    cs = cand_score + (size_t)bc * KPAD;
    const unsigned* ci = cand_idx   + (size_t)bc * KPAD;
    const float4*   cb = reinterpret_cast<const float4*>(cand_box) + (size_t)bc * KPAD;

    for (int s = t; s < KPAD; s += 256) {
        float4 bx = cb[s];
        candBoxL[s] = bx;
        areaArr[s]  = (bx.z - bx.x) * (bx.w - bx.y);
    }
    for (int wd = t; wd < KPAD * 16; wd += 256) supMat[wd] = 0u;
    __syncthreads();

    int wave = t >> 5, lane = t & 31;
    int half = lane >> 4;        // ISA A layout: lanes16-31 carry K2/K3 (zeroed)
    int lan15 = lane & 15;

    for (int tile = wave; tile < 32 * 32; tile += 8) {   // uniform per wave: EXEC all-ones
        int ti = tile >> 5, tj = tile & 31;
        v2f a, bbv;
        a.x   = half ? 0.f : areaArr[ti * 16 + lan15];   // A col K0 = area_i
        a.y   = half ? 0.f : 1.f;                        // A col K1 = ones
        bbv.x = half ? 0.f : 1.f;                        // B row K0 = ones
        bbv.y = half ? 0.f : areaArr[tj * 16 + lan15];   // B row K1 = area_j
        v8f c = {0.f, 0.f, 0.f, 0.f, 0.f, 0.f, 0.f, 0.f};
        v8f d = __builtin_amdgcn_wmma_f32_16x16x4_f32(
            false, a, false, bbv, (short)0, c, false, false);

        int n = tj * 16 + lan15;
        float4 bj = candBoxL[n];
        int mbase = ti * 16 + half * 8;                  // D: VGPR r -> row r+8*half
        #pragma unroll
        for (int r = 0; r < 8; ++r) {
            int m = mbase + r;
            float4 bi = candBoxL[m];
            float iw = fminf(bi.z, bj.z) - fmaxf(bi.x, bj.x);
            float ih = fminf(bi.w, bj.w) - fmaxf(bi.y, bj.y);
            float inter = fmaxf(iw, 0.f) * fmaxf(ih, 0.f);
            float uni = d[r] - inter;                    // (area_i+area_j) from WMMA
            if (inter > NMS_TH * uni)                    // iou > TH, union > 0
                atomicOr(&supMat[m * 16 + (n >> 5)], 1u << (n & 31));
        }
    }
    __syncthreads();
    if (wave != 0) return;

    // wave32 greedy: 16 slots/lane, keys = (score_bits<<32)|~prior_idx
    unsigned long long key[16];
    unsigned active = 0;
    #pragma unroll
    for (int k = 0; k < 16; ++k) {
        int s = lane * 16 + k;
        float sv = cs[s];
        if (sv > CONF_TH) {
            key[k] = ((unsigned long long)fbits(sv) << 32) | (unsigned)(~ci[s]);
            active |= (1u << k);
        } else key[k] = 0ull;
    }

    volatile unsigned long long* vK  = redK;
    volatile int*                vS  = redS;
    volatile unsigned long long* vgK = &gKey;
    volatile int*                vgS = &gSlot;
    float* orow = out + ((size_t)(b * NCLS + (cl + 1))) * TOPK * 5;

    for (int iter = 0; iter < TOPK; ++iter) {
        unsigned long long best = 0ull; int bslot = 0;
        #pragma unroll
        for (int k = 0; k < 16; ++k)
            if ((active >> k) & 1u)
                if (key[k] > best) { best = key[k]; bslot = lane * 16 + k; }
        vK[lane] = best; vS[lane] = bslot;
        if (lane == 0) {
            unsigned long long g = 0ull; int gs = 0;
            for (int l = 0; l < 32; ++l) {
                unsigned long long kk = vK[l];
                if (kk > g) { g = kk; gs = vS[l]; }
            }
            *vgK = g; *vgS = gs;
        }
        unsigned long long g = *vgK; int gs = *vgS;    // lockstep wave32 broadcast
        if (g == 0ull) break;                          // remaining rows stay zero
        if (lane == 0) {
            float4 bx = candBoxL[gs];
            float* row = orow + (size_t)iter * 5;
            row[0] = __uint_as_float((unsigned)(g >> 32));
            row[1] = bx.x; row[2] = bx.y; row[3] = bx.z; row[4] = bx.w;
        }
        unsigned wbits = supMat[gs * 16 + (lane >> 1)];
        unsigned h16 = (wbits >> ((lane & 1) * 16)) & 0xFFFFu;
        active &= ~h16;
        if ((gs >> 4) == lane) active &= ~(1u << (gs & 15));
    }
}

// --------------------------------------------------------------------- launcher
extern "C" void kernel_launch(void* const* d_in, const int* in_sizes, int n_in,
                              void* d_out, int out_size, void* d_ws, size_t ws_size,
                              hipStream_t stream) {
    (void)in_sizes; (void)n_in;
    const float* arm_loc  = (const float*)d_in[0];
    const float* arm_conf = (const float*)d_in[1];
    const float* odm_loc  = (const float*)d_in[2];
    const float* odm_conf = (const float*)d_in[3];
    const float* priors   = (const float*)d_in[4];
    float* out = (float*)d_out;

    char* ws = (char*)d_ws;
    size_t off = 0;
    float* boxes  = (float*)(ws + off); off += (size_t)BATCH * P_NUM * 4 * sizeof(float);
    float* conf_t = (float*)(ws + off); off += (size_t)BATCH * 20 * P_NUM * sizeof(float);
    float* cand_score = (float*)(ws + off); off += (size_t)BATCH * 20 * KPAD * sizeof(float);
    unsigned* cand_idx = (unsigned*)(ws + off); off += (size_t)BATCH * 20 * KPAD * sizeof(unsigned);
    float* cand_box = (float*)(ws + off); off += (size_t)BATCH * 20 * KPAD * 4 * sizeof(float);
    if (ws_size < off) return;   // scratch insufficient: ~58 MB required

    k_zero<<<(out_size + 255) / 256, 256, 0, stream>>>(out, out_size);
    k_decode<<<(BATCH * P_NUM + 255) / 256, 256, 0, stream>>>(
        arm_loc, arm_conf, odm_loc, odm_conf, priors, boxes, conf_t);
    k_select<<<BATCH * 20, 256, 0, stream>>>(conf_t, boxes, cand_score, cand_idx, cand_box);
    k_nms<<<BATCH * 20, 256, 0, stream>>>(cand_score, cand_idx, cand_box, out);
}